// MassMatrix_24515673325989
// MI455X (gfx1250) — compile-verified
//
#include <hip/hip_runtime.h>
#include <hip/hip_bf16.h>

// Problem constants (from reference): N=1024 matrices of K=128.
#define NN 1024
#define KK 128
#define KK2 (KK * KK)            // 16384
#define WCOLS (NN * 16)          // big-GEMM N dimension = 16384

typedef float v2f __attribute__((ext_vector_type(2)));
typedef float v8f __attribute__((ext_vector_type(8)));

// ---------------------------------------------------------------------------
// 1) Batched in-place Gauss-Jordan inversion of 128x128 (no pivoting; omega is
//    a well-conditioned perturbation of I). One block per matrix, thread = row.
// ---------------------------------------------------------------------------
__global__ void __launch_bounds__(128) gj_invert_kernel(
    const float* __restrict__ omega, float* __restrict__ inv) {
  __shared__ float A[KK * KK];               // 64 KB LDS
  const int n = blockIdx.x;
  const int r = threadIdx.x;                  // my row
  const float* src = omega + (size_t)n * KK2;

  #pragma unroll 4
  for (int j = 0; j < KK; ++j) A[r * KK + j] = src[r * KK + j];
  __syncthreads();

  for (int p = 0; p < KK; ++p) {
    const float d = 1.0f / A[p * KK + p];
    // Phase 1: eliminate column p from every row r != p using the UNSCALED
    // pivot row (factor f*d folded in). Row p is untouched here.
    if (r != p) {
      const float f = A[r * KK + p] * d;
      #pragma unroll 4
      for (int j = 0; j < KK; ++j) {
        if (j != p) A[r * KK + j] -= f * A[p * KK + j];
      }
      A[r * KK + p] = -f;
    }
    __syncthreads();
    // Phase 2: scale pivot row cooperatively (thread r owns element (p,r)).
    A[p * KK + r] = (r == p) ? d : A[p * KK + r] * d;
    __syncthreads();
  }

  float* dst = inv + (size_t)n * KK2;
  #pragma unroll 4
  for (int j = 0; j < KK; ++j) dst[r * KK + j] = A[r * KK + j];
}

// ---------------------------------------------------------------------------
// 2) beta_nd = beta with zeroed diagonal (hoists all masking out of the
//    dominant GEMM's inner loop).
// ---------------------------------------------------------------------------
__global__ void __launch_bounds__(256) mask_diag_kernel(
    const float* __restrict__ beta, float* __restrict__ bnd) {
  const size_t idx = (size_t)blockIdx.x * blockDim.x + threadIdx.x;
  const int i = (int)(idx >> 10);
  const int j = (int)(idx & 1023);
  bnd[idx] = (i == j) ? 0.0f : beta[idx];
}

// ---------------------------------------------------------------------------
// 3) col_sum[j] = sum_i beta_nd[i][j]
// ---------------------------------------------------------------------------
__global__ void __launch_bounds__(256) colsum_kernel(
    const float* __restrict__ bnd, float* __restrict__ cs) {
  const int j = blockIdx.x * blockDim.x + threadIdx.x;
  if (j >= NN) return;
  float s = 0.0f;
  for (int i = 0; i < NN; ++i) s += bnd[(size_t)i * NN + j];
  cs[j] = s;
}

// ---------------------------------------------------------------------------
// 4) Batched 128x128x128 GEMM via V_WMMA_F32_16X16X4_F32.
//    C[b] = op(A[b]) * B[b], op = transpose if TRANSA.
//    One block (8 waves) per matrix. B[b] staged in LDS in a k-pair
//    INTERLEAVED layout: element (k,n) -> Bs[k>>1][2n + (k&1)], so each
//    B fragment {B[k0][n], B[k0+1][n]} (k0 even) is ONE aligned ds_load_b64
//    into an even VGPR pair -- no shuffle movs.
//    Each wave owns a 16x128 strip -> 8 accumulators, 8 WMMA per A-fragment.
//    FUSE adds the elementwise epilogue (Lp + colsum*Lq + obs) at store time.
//    Fragment layouts per CDNA5 ISA 7.12.2:
//      A 16x4 : lane<16 -> {K=k0,k0+1} of row M=lane%16 ; lane>=16 -> {k0+2,k0+3}
//      B 4x16 : vgpr0 = rows K=k0 / k0+2, vgpr1 = rows k0+1 / k0+3, N = lane%16
//      C/D    : vgpr r = row r (lanes 0-15) and row r+8 (lanes 16-31)
// ---------------------------------------------------------------------------
#define BP2 260  // interleaved pair-row stride (2*128+4): 16B-aligned

template <bool TRANSA, bool FUSE>
__global__ void __launch_bounds__(256) batched_gemm128_kernel(
    const float* __restrict__ Ag, const float* __restrict__ Bg,
    float* __restrict__ Cg, const float* __restrict__ pp,
    const float* __restrict__ pq, const float* __restrict__ obs,
    const float* __restrict__ cs) {
  __shared__ float Bs[KK / 2][BP2];           // 65 KB LDS
  const int lane = threadIdx.x & 31;
  const int wave = threadIdx.x >> 5;
  const size_t b = blockIdx.x;

  const float* A = Ag + b * (size_t)KK2;
  const float* B = Bg + b * (size_t)KK2;

  // Cooperative interleaved stage of B[b]: 4 threads per row-pair.
  {
    const int kp = threadIdx.x >> 2;          // 0..63 row pair
    const int c0 = (threadIdx.x & 3) * 32;    // 32 columns per thread
    #pragma unroll
    for (int v = 0; v < 8; ++v) {
      const int nn = c0 + v * 4;
      const float4 e = *(const float4*)(B + (size_t)(2 * kp) * KK + nn);
      const float4 o = *(const float4*)(B + (size_t)(2 * kp + 1) * KK + nn);
      *(float4*)&Bs[kp][2 * nn] = make_float4(e.x, o.x, e.y, o.y);
      *(float4*)&Bs[kp][2 * nn + 4] = make_float4(e.z, o.z, e.w, o.w);
    }
  }
  __syncthreads();

  const int half = lane >> 4;                 // 0 or 1
  const int l16 = lane & 15;
  const int kb = half * 2;                    // even -> k0 always even
  const int m0 = wave * 16;                   // this wave's 16-row strip
  const float csb = FUSE ? cs[b] : 0.0f;

  v8f acc[8] = {};
  #pragma unroll 4
  for (int kk = 0; kk < KK; kk += 4) {
    const int k0 = kk + kb;
    v2f a;
    if (TRANSA) {
      a.x = A[(size_t)k0 * KK + (m0 + l16)];
      a.y = A[(size_t)(k0 + 1) * KK + (m0 + l16)];
    } else {
      const float* ap = A + (size_t)(m0 + l16) * KK + k0;
      a.x = ap[0];
      a.y = ap[1];
    }
    #pragma unroll
    for (int t = 0; t < 8; ++t) {
      const v2f bv = *(const v2f*)&Bs[k0 >> 1][2 * (t * 16 + l16)];
      acc[t] = __builtin_amdgcn_wmma_f32_16x16x4_f32(false, a, false, bv,
                                                     (short)0, acc[t], false,
                                                     false);
    }
  }

  #pragma unroll
  for (int t = 0; t < 8; ++t) {
    const int ncol = t * 16 + l16;
    #pragma unroll
    for (int r = 0; r < 8; ++r) {
      const int rg = m0 + r + half * 8;
      const size_t idx = b * (size_t)KK2 + (size_t)rg * KK + ncol;
      float v = acc[t][r];
      if (FUSE) v += pp[idx] + csb * pq[idx] + obs[rg * KK + ncol];
      Cg[idx] = v;
    }
  }
}

// ---------------------------------------------------------------------------
// 5) W = beta_nd @ S_flat : (1024x1024) * (1024x16384).
//    Block = 8 waves covering 128 rows x 64 cols; the shared 32x64 k-chunk of
//    S is staged in LDS with the same k-pair interleaved layout; each wave
//    runs 4 accumulators. A-fragments are plain global b64 loads of beta_nd.
// ---------------------------------------------------------------------------
#define MP2 132  // interleaved pair-row stride (2*64+4): 16B-aligned

__global__ void __launch_bounds__(256) beta_mix_gemm_kernel(
    const float* __restrict__ bnd, const float* __restrict__ S,
    float* __restrict__ W) {
  __shared__ float Bs[16][MP2];               // 8.25 KB
  const int lane = threadIdx.x & 31;
  const int wave = threadIdx.x >> 5;
  const int mb = blockIdx.x >> 8;             // 8 row super-blocks of 128
  const int nb = blockIdx.x & 255;            // 256 col strips of 64
  const int n0 = nb * 64;

  const int half = lane >> 4;
  const int l16 = lane & 15;
  const int kb = half * 2;
  const int m0 = mb * 128 + wave * 16;
  const int row = m0 + l16;                   // the beta row this lane feeds

  const int kp = threadIdx.x >> 4;            // loader: pair-row 0..15
  const int c0 = (threadIdx.x & 15) * 4;      // 4 columns

  v8f acc[4] = {};
  for (int kc = 0; kc < NN; kc += 32) {
    __syncthreads();
    {  // interleaved stage of S[kc:kc+32, n0:n0+64]
      const float4 e =
          *(const float4*)(S + (size_t)(kc + 2 * kp) * WCOLS + n0 + c0);
      const float4 o =
          *(const float4*)(S + (size_t)(kc + 2 * kp + 1) * WCOLS + n0 + c0);
      *(float4*)&Bs[kp][2 * c0] = make_float4(e.x, o.x, e.y, o.y);
      *(float4*)&Bs[kp][2 * c0 + 4] = make_float4(e.z, o.z, e.w, o.w);
    }
    __syncthreads();
    #pragma unroll
    for (int k4 = 0; k4 < 32; k4 += 4) {
      const int k0 = k4 + kb;                 // even
      v2f a;
      const float* ap = bnd + (size_t)row * NN + kc + k0;
      a.x = ap[0];
      a.y = ap[1];
      #pragma unroll
      for (int t = 0; t < 4; ++t) {
        const v2f bv = *(const v2f*)&Bs[k0 >> 1][2 * (t * 16 + l16)];
        acc[t] = __builtin_amdgcn_wmma_f32_16x16x4_f32(false, a, false, bv,
                                                       (short)0, acc[t], false,
                                                       false);
      }
    }
  }

  #pragma unroll
  for (int t = 0; t < 4; ++t) {
    #pragma unroll
    for (int r = 0; r < 8; ++r) {
      W[(size_t)(m0 + r + half * 8) * WCOLS + n0 + t * 16 + l16] = acc[t][r];
    }
  }
}

// ---------------------------------------------------------------------------
// Host-side orchestration.
// Workspace layout (needs 132 MB + 4 KB):
//   B0  = omega_inv   [64 MB]
//   B1  = rotating    [64 MB]
//   bnd = beta_nd     [4 MB]
//   cs  = column sums [4 KB]
// Schedule (each step reads/writes disjoint buffers):
//   1. inv(omega)                          -> B0
//   2. beta_nd = mask(beta)                -> bnd
//   3. colsum(beta_nd)                     -> cs
//   4. T1 = Lq @ Om                        -> d_out
//   5. S  = Om^T @ T1(d_out)               -> B1
//   6. W  = beta_nd @ S(B1)                -> d_out
//   7. T2 = W(d_out) @ OmInv(B0)           -> B1
//   8. out = OmInv^T(B0) @ T2(B1)
//            + Lp + cs*Lq + obs  (fused)   -> d_out
// ---------------------------------------------------------------------------
extern "C" void kernel_launch(void* const* d_in, const int* in_sizes, int n_in,
                              void* d_out, int out_size, void* d_ws,
                              size_t ws_size, hipStream_t stream) {
  const float* beta = (const float*)d_in[0];   // (N,N)
  const float* omega = (const float*)d_in[1];  // (N,K,K)
  const float* pp = (const float*)d_in[2];     // precision_p (N,K,K)
  const float* pq = (const float*)d_in[3];     // precision_q (N,K,K)
  const float* obs = (const float*)d_in[4];    // obs_precision (K,K)

  float* out = (float*)d_out;                  // (N,K,K)
  const size_t bigElems = (size_t)NN * KK2;    // 16,777,216 floats
  float* B0 = (float*)d_ws;                    // omega_inv
  float* B1 = B0 + bigElems;                   // rotating temp
  float* bnd = B1 + bigElems;                  // beta with zeroed diagonal
  float* cs = bnd + (size_t)NN * NN;           // 1024 floats

  const float* nullf = nullptr;
  const dim3 blk256(256), blk128(128);
  const dim3 gridInv(NN);                 // 1024 blocks
  const dim3 gridBat(NN);                 // one block per matrix
  const dim3 gridMix(8 * 256);            // 8 row-super x 256 col-strips
  const dim3 gridMask((NN * NN) / 256);   // 4096 blocks
  const dim3 gridCs(NN / 256);

  // 1) omega_inv
  gj_invert_kernel<<<gridInv, blk128, 0, stream>>>(omega, B0);
  // 2) beta_nd
  mask_diag_kernel<<<gridMask, blk256, 0, stream>>>(beta, bnd);
  // 3) column sums
  colsum_kernel<<<gridCs, blk256, 0, stream>>>(bnd, cs);
  // 4) T1 = Lq @ Om
  batched_gemm128_kernel<false, false><<<gridBat, blk256, 0, stream>>>(
      pq, omega, out, nullf, nullf, nullf, nullf);
  // 5) S = Om^T @ T1
  batched_gemm128_kernel<true, false><<<gridBat, blk256, 0, stream>>>(
      omega, out, B1, nullf, nullf, nullf, nullf);
  // 6) W = beta_nd @ S
  beta_mix_gemm_kernel<<<gridMix, blk256, 0, stream>>>(bnd, B1, out);
  // 7) T2 = W @ OmInv
  batched_gemm128_kernel<false, false><<<gridBat, blk256, 0, stream>>>(
      out, B0, B1, nullf, nullf, nullf, nullf);
  // 8) cross + fused epilogue
  batched_gemm128_kernel<true, true><<<gridBat, blk256, 0, stream>>>(
      B0, B1, out, pp, pq, obs, cs);
}